// GCNEncoder_87316685127964
// MI455X (gfx1250) — compile-verified
//
#include <hip/hip_runtime.h>
#include <math.h>

// ---------------------------------------------------------------------------
// GCN encoder for MI455X (gfx1250, wave32).
//   GEMMs via V_WMMA_F32_16X16X4_F32 (native fp32 matrix path, keeps accuracy)
//     - one wave owns a 16-row strip across all output columns: A fragment
//       loaded once per K-step (global_load_b64) and reused by COUT/16 WMMAs
//     - W staged in LDS in K-pair-packed float2 layout, so each B fragment is
//       one ds_load_b64 straight into an even-aligned VGPR pair (no repacking)
//   Edge aggregation via wave-per-edge gather + global_atomic_add_f32 scatter
//   (feature matrix = 51 MB, fits the 192 MB L2 -> atomics stay on-chip).
// ---------------------------------------------------------------------------

typedef __attribute__((ext_vector_type(2))) float v2f;
typedef __attribute__((ext_vector_type(8))) float v8f;

#define N_NODES 100000
#define N_EDGES 1600000
#define EPSBN   1e-5f

// ---------------------------------------------------------------------------
// Degree / norm precompute
// ---------------------------------------------------------------------------
__global__ void k_init_deg(float* __restrict__ deg) {
    int i = blockIdx.x * blockDim.x + threadIdx.x;
    if (i < N_NODES) deg[i] = 1.0f;             // self-loop counts as 1
}

__global__ void k_edge_deg(const int* __restrict__ dst, float* __restrict__ deg) {
    int e = blockIdx.x * blockDim.x + threadIdx.x;
    if (e < N_EDGES) atomicAdd(&deg[dst[e]], 1.0f);
}

__global__ void k_dis(const float* __restrict__ deg, float* __restrict__ dis) {
    int i = blockIdx.x * blockDim.x + threadIdx.x;
    if (i < N_NODES) dis[i] = rsqrtf(deg[i]);   // deg >= 1 always
}

// ---------------------------------------------------------------------------
// GEMM: out[N_NODES x COUT] = A[N_NODES x CIN] * W[CIN x COUT]
// Block = 256 threads = 8 waves. W staged into LDS once, K-pair packed:
//   Wlds[kp*COUT + c] = { W[2kp][c], W[2kp+1][c] }   (float2)
// Each wave computes a 16 x COUT strip: per K-step (K=4) the A fragment is
// loaded once from global, B fragments are single ds_load_b64 reads.
// A layout per ISA: lanes 0-15 hold rows M=0..15 with K={k0,k0+1},
//                   lanes 16-31 hold same rows with K={k0+2,k0+3}.
// C/D layout: VGPR i -> (M = i + 8*(lane>=16), N = lane%16).
// ---------------------------------------------------------------------------
template <int CIN, int COUT>
__global__ __launch_bounds__(256) void k_gemm_wmma(const float* __restrict__ A,
                                                   const float* __restrict__ W,
                                                   float* __restrict__ out) {
    constexpr int TN = COUT / 16;                 // column tiles per wave
    __shared__ v2f Wlds[(CIN / 2) * COUT];

    // --- cooperative W -> LDS, K-pair packed (coalesced row reads) ---
    for (int idx = threadIdx.x; idx < (CIN / 2) * COUT; idx += 256) {
        int kp = idx / COUT;                      // compile-time COUT -> shift
        int c  = idx % COUT;
        v2f w;
        w.x = W[(size_t)(2 * kp)     * COUT + c];
        w.y = W[(size_t)(2 * kp + 1) * COUT + c];
        Wlds[idx] = w;
    }
    __syncthreads();

    const int lane  = threadIdx.x & 31;
    const int wave  = threadIdx.x >> 5;
    const int tm    = blockIdx.x * 8 + wave;      // 16-row strip index
    if (tm >= N_NODES / 16) return;               // 100000 % 16 == 0

    const int l16   = lane & 15;
    const int khalf = (lane >> 4) << 1;           // 0 or 2
    const int row   = tm * 16 + l16;              // A row owned by this lane

    const float* __restrict__ Arow = A + (size_t)row * CIN;

    v8f acc[TN];
#pragma unroll
    for (int t = 0; t < TN; ++t) acc[t] = (v8f){};

#pragma unroll 4
    for (int k0 = 0; k0 < CIN; k0 += 4) {
        v2f a = *(const v2f*)(Arow + k0 + khalf);             // contiguous K pair
        const v2f* __restrict__ B0 = &Wlds[((k0 + khalf) >> 1) * COUT + l16];
#pragma unroll
        for (int t = 0; t < TN; ++t) {
            v2f b = B0[t * 16];                               // one ds_load_b64
            acc[t] = __builtin_amdgcn_wmma_f32_16x16x4_f32(
                        /*neg_a=*/false, a, /*neg_b=*/false, b,
                        /*c_mod=*/(short)0, acc[t],
                        /*reuse_a=*/false, /*reuse_b=*/false);
        }
    }

    const int rbase = tm * 16 + ((lane >> 4) << 3);           // +8 for hi lanes
#pragma unroll
    for (int t = 0; t < TN; ++t) {
        float* __restrict__ O = out + (size_t)rbase * COUT + t * 16 + l16;
#pragma unroll
        for (int i = 0; i < 8; ++i)
            O[(size_t)i * COUT] = acc[t][i];
    }
}

// ---------------------------------------------------------------------------
// Aggregation: out initialized with the self-loop term h[i]*dis[i]^2, then
// edges scatter h[src]*dis[src]*dis[dst] into out[dst] with f32 atomics.
// ---------------------------------------------------------------------------
template <int C>
__global__ void k_selfloop_init(const float* __restrict__ h,
                                const float* __restrict__ dis,
                                float* __restrict__ out) {
    long long i = (long long)blockIdx.x * blockDim.x + threadIdx.x;
    if (i >= (long long)N_NODES * C) return;
    int node = (int)(i / C);
    float d = dis[node];
    out[i] = h[i] * d * d;
}

template <int C>  // C must be 64 or 128; one wave per edge
__global__ __launch_bounds__(256) void k_edge_aggregate(const int* __restrict__ src,
                                                        const int* __restrict__ dst,
                                                        const float* __restrict__ dis,
                                                        const float* __restrict__ h,
                                                        float* __restrict__ out) {
    const int lane = threadIdx.x & 31;
    long long e = (long long)blockIdx.x * 8 + (threadIdx.x >> 5);
    if (e >= N_EDGES) return;
    const int s = src[e];
    const int d = dst[e];
    const float nrm = dis[s] * dis[d];
    constexpr int V = C / 32;                    // floats per lane (4 or 2)
    const float* __restrict__ hp = h   + (size_t)s * C + lane * V;
    float* __restrict__       op = out + (size_t)d * C + lane * V;
    if constexpr (V == 4) {
        float4 v = *(const float4*)hp;
        atomicAdd(op + 0, v.x * nrm);
        atomicAdd(op + 1, v.y * nrm);
        atomicAdd(op + 2, v.z * nrm);
        atomicAdd(op + 3, v.w * nrm);
    } else {
        float2 v = *(const float2*)hp;
        atomicAdd(op + 0, v.x * nrm);
        atomicAdd(op + 1, v.y * nrm);
    }
}

// ---------------------------------------------------------------------------
// BatchNorm (training-mode, biased variance) + ReLU.
// Per-column constants (conv bias) cancel inside BN, so b1/b2 are skipped.
// ---------------------------------------------------------------------------
__global__ void k_zero_stats(float* __restrict__ stats) {
    int i = threadIdx.x;                          // 256 threads
    stats[i] = 0.0f;
}

__global__ __launch_bounds__(128) void k_col_stats(const float* __restrict__ x,
                                                   float* __restrict__ stats,
                                                   int rows_per_block) {
    const int c  = threadIdx.x;                   // column 0..127
    int r0 = blockIdx.x * rows_per_block;
    int r1 = r0 + rows_per_block;
    if (r1 > N_NODES) r1 = N_NODES;
    float s = 0.0f, q = 0.0f;
    for (int r = r0; r < r1; ++r) {
        float v = x[(size_t)r * 128 + c];         // coalesced across threads
        s += v;
        q += v * v;
    }
    atomicAdd(&stats[c],       s);
    atomicAdd(&stats[128 + c], q);
}

__global__ void k_bn_coeffs(const float* __restrict__ stats,
                            const float* __restrict__ gamma,
                            const float* __restrict__ beta,
                            float* __restrict__ coeff) {   // [0:128]=scale [128:256]=shift
    int c = threadIdx.x;                          // 128 threads
    float mean = stats[c] * (1.0f / N_NODES);
    float var  = stats[128 + c] * (1.0f / N_NODES) - mean * mean;
    float sc   = gamma[c] * rsqrtf(var + EPSBN);
    coeff[c]       = sc;
    coeff[128 + c] = beta[c] - mean * sc;
}

__global__ void k_bn_relu(float* __restrict__ x, const float* __restrict__ coeff) {
    long long i = (long long)blockIdx.x * blockDim.x + threadIdx.x;
    if (i >= (long long)N_NODES * 128) return;
    int c = (int)(i & 127);
    float v = x[i] * coeff[c] + coeff[128 + c];
    x[i] = v > 0.0f ? v : 0.0f;
}

// Final-layer output init: self-loop term + conv bias b3 (no BN follows).
__global__ void k_out_init(const float* __restrict__ h,
                           const float* __restrict__ dis,
                           const float* __restrict__ b3,
                           float* __restrict__ out) {
    long long i = (long long)blockIdx.x * blockDim.x + threadIdx.x;
    if (i >= (long long)N_NODES * 64) return;
    int node = (int)(i >> 6);
    int c    = (int)(i & 63);
    float d = dis[node];
    out[i] = h[i] * d * d + b3[c];
}

// ---------------------------------------------------------------------------
// Host-side orchestration
// ---------------------------------------------------------------------------
extern "C" void kernel_launch(void* const* d_in, const int* in_sizes, int n_in,
                              void* d_out, int out_size, void* d_ws, size_t ws_size,
                              hipStream_t stream) {
    (void)in_sizes; (void)n_in; (void)out_size; (void)ws_size;

    const float* x    = (const float*)d_in[0];        // [N,128]
    const int*   ei   = (const int*)d_in[1];          // [2,E] (int32)
    const float* W1   = (const float*)d_in[2];        // [128,128]
    const float* g1   = (const float*)d_in[4];
    const float* bt1  = (const float*)d_in[5];
    const float* W2   = (const float*)d_in[6];        // [128,128]
    const float* g2   = (const float*)d_in[8];
    const float* bt2  = (const float*)d_in[9];
    const float* W3   = (const float*)d_in[10];       // [128,64]
    const float* b3   = (const float*)d_in[11];
    float* out = (float*)d_out;                       // [N,64]

    const int* src = ei;                              // row 0
    const int* dst = ei + N_EDGES;                    // row 1

    // Workspace partition (floats)
    float* ws    = (float*)d_ws;
    float* deg   = ws;                                // N
    float* dis   = deg + N_NODES;                     // N
    float* bufA  = dis + N_NODES;                     // N*128
    float* bufB  = bufA + (size_t)N_NODES * 128;      // N*128
    float* stats = bufB + (size_t)N_NODES * 128;      // 256
    float* coeff = stats + 256;                       // 256

    const int T = 256;
    const int gN    = (N_NODES + T - 1) / T;
    const int gE    = (N_EDGES + T - 1) / T;
    const int gE8   = (N_EDGES + 7) / 8;              // wave-per-edge
    const int gF128 = (int)(((long long)N_NODES * 128 + T - 1) / T);
    const int gF64  = (int)(((long long)N_NODES * 64  + T - 1) / T);
    const int gGemm = (N_NODES / 16 + 7) / 8;         // 16-row strips / 8 waves
    const int rowsPerBlk = 500;
    const int gStats = (N_NODES + rowsPerBlk - 1) / rowsPerBlk;

    // --- degree / symmetric norm ---
    k_init_deg<<<gN, T, 0, stream>>>(deg);
    k_edge_deg<<<gE, T, 0, stream>>>(dst, deg);
    k_dis<<<gN, T, 0, stream>>>(deg, dis);

    // --- layer 1: GEMM -> aggregate -> BN+ReLU ---
    k_gemm_wmma<128, 128><<<gGemm, T, 0, stream>>>(x, W1, bufA);
    k_selfloop_init<128><<<gF128, T, 0, stream>>>(bufA, dis, bufB);
    k_edge_aggregate<128><<<gE8, T, 0, stream>>>(src, dst, dis, bufA, bufB);
    k_zero_stats<<<1, 256, 0, stream>>>(stats);
    k_col_stats<<<gStats, 128, 0, stream>>>(bufB, stats, rowsPerBlk);
    k_bn_coeffs<<<1, 128, 0, stream>>>(stats, g1, bt1, coeff);
    k_bn_relu<<<gF128, T, 0, stream>>>(bufB, coeff);

    // --- layer 2: GEMM -> aggregate -> BN+ReLU ---
    k_gemm_wmma<128, 128><<<gGemm, T, 0, stream>>>(bufB, W2, bufA);
    k_selfloop_init<128><<<gF128, T, 0, stream>>>(bufA, dis, bufB);
    k_edge_aggregate<128><<<gE8, T, 0, stream>>>(src, dst, dis, bufA, bufB);
    k_zero_stats<<<1, 256, 0, stream>>>(stats);
    k_col_stats<<<gStats, 128, 0, stream>>>(bufB, stats, rowsPerBlk);
    k_bn_coeffs<<<1, 128, 0, stream>>>(stats, g2, bt2, coeff);
    k_bn_relu<<<gF128, T, 0, stream>>>(bufB, coeff);

    // --- layer 3: GEMM -> aggregate (+b3) into d_out ---
    k_gemm_wmma<128, 64><<<gGemm, T, 0, stream>>>(bufB, W3, bufA);
    k_out_init<<<gF64, T, 0, stream>>>(bufA, dis, b3, out);
    k_edge_aggregate<64><<<gE8, T, 0, stream>>>(src, dst, dis, bufA, out);
}